// GraphSAGE_46033459479145
// MI455X (gfx1250) — compile-verified
//
#include <hip/hip_runtime.h>
#include <hip/hip_bf16.h>

typedef __attribute__((ext_vector_type(2))) float v2f;
typedef __attribute__((ext_vector_type(8))) float v8f;

#define N_NODES 100000
#define N_EDGES 1600000
#define D 128

// ---------------------------------------------------------------------------
// Edge scatter: msg[dst][d] += h[src][d]. 128 threads per edge, 1 float each.
// unsafeAtomicAdd lowers to global_atomic_add_f32 (no-return, STOREcnt).
// ---------------------------------------------------------------------------
__global__ void sage_scatter(const float* __restrict__ h,
                             const int* __restrict__ esrc,
                             const int* __restrict__ edst,
                             float* __restrict__ msg) {
    unsigned gid = blockIdx.x * blockDim.x + threadIdx.x;
    unsigned e = gid >> 7;       // edge index
    unsigned d = gid & 127u;     // feature index
    if (e >= N_EDGES) return;
    int s = esrc[e];
    int t = edst[e];
    float v = h[(size_t)s * D + d];
    unsafeAtomicAdd(&msg[(size_t)t * D + d], v);
}

// ---------------------------------------------------------------------------
// Fused SAGE GEMM: out[16 x 128 tile] = Aself@Wself + (Amsg*inv_deg)@Wneigh + b
// One block = 8 waves; wave w computes the 16x16 tile at columns [16w,16w+16).
// fp32 WMMA 16x16x4, K swept 0..128 in steps of 4, twice (self + neigh).
//
// ISA VGPR layouts (CDNA5 ISA):
//   A 16x4:  lane l<16 -> row=l, K={k0,k0+1}; lane 16+l -> row=l, K={k0+2,k0+3}
//   B 4x16:  lane l<16 -> col=l, K={k0,k0+1}; lane 16+l -> col=l, K={k0+2,k0+3}
//   C/D:     vgpr r, lane half hf -> element (M = r + 8*hf, N = lane&15)
// ---------------------------------------------------------------------------
__global__ void sage_gemm(const float* __restrict__ Aself,
                          const float* __restrict__ Amsg,
                          const float* __restrict__ Wself,
                          const float* __restrict__ Wneigh,
                          const float* __restrict__ bias,
                          const int* __restrict__ indeg,
                          float* __restrict__ out) {
    const int tile = blockIdx.x;            // 16 node rows per block
    const int wave = threadIdx.x >> 5;      // 0..7
    const int lane = threadIdx.x & 31;
    const int half = lane >> 4;             // 0 or 1 -> which K pair / which M half
    const int l    = lane & 15;

    const int row = tile * 16 + l;          // A row this lane feeds
    const int col = wave * 16 + l;          // output column this lane holds

    const float inv = 1.0f / fmaxf((float)indeg[row], 1.0f);

    const float* arow_s = Aself + (size_t)row * D + half * 2;
    const float* arow_m = Amsg  + (size_t)row * D + half * 2;
    const float* bcol_s = Wself  + (size_t)(half * 2) * D + col;
    const float* bcol_m = Wneigh + (size_t)(half * 2) * D + col;

    v8f c = {};

    // ---- self term: Aself @ Wself ----
#pragma unroll
    for (int k0 = 0; k0 < D; k0 += 4) {
        v2f a = *(const v2f*)(arow_s + k0);        // contiguous pair (b64 load)
        v2f b;
        b.x = bcol_s[(size_t)k0 * D];              // W[k0+2h  ][col]
        b.y = bcol_s[(size_t)(k0 + 1) * D];        // W[k0+2h+1][col]
        c = __builtin_amdgcn_wmma_f32_16x16x4_f32(false, a, false, b,
                                                  (short)0, c, false, false);
    }

    // ---- neighbor term: (Amsg * inv_deg) @ Wneigh ----
#pragma unroll
    for (int k0 = 0; k0 < D; k0 += 4) {
        v2f a = *(const v2f*)(arow_m + k0);
        a.x *= inv;
        a.y *= inv;
        v2f b;
        b.x = bcol_m[(size_t)k0 * D];
        b.y = bcol_m[(size_t)(k0 + 1) * D];
        c = __builtin_amdgcn_wmma_f32_16x16x4_f32(false, a, false, b,
                                                  (short)0, c, false, false);
    }

    const float bv = bias[col];
    // store D tile: vgpr r -> node row (tile*16 + r + 8*half), column col
    float* obase = out + ((size_t)tile * 16 + half * 8) * D + col;
#pragma unroll
    for (int r = 0; r < 8; ++r) {
        obase[(size_t)r * D] = c[r] + bv;
    }
}

// ---------------------------------------------------------------------------
// In-place LayerNorm + ReLU over each 128-wide row. One block per node.
// ---------------------------------------------------------------------------
__global__ void ln_relu(float* __restrict__ h,
                        const float* __restrict__ g,
                        const float* __restrict__ b) {
    const int node = blockIdx.x;
    const int t = threadIdx.x;          // 0..127
    float x = h[(size_t)node * D + t];

    __shared__ float red0[4];
    __shared__ float red1[4];

    float s = x;
#pragma unroll
    for (int o = 16; o > 0; o >>= 1) s += __shfl_xor(s, o, 32);
    if ((t & 31) == 0) red0[t >> 5] = s;
    __syncthreads();
    const float mu = (red0[0] + red0[1] + red0[2] + red0[3]) * (1.0f / D);

    const float dx = x - mu;
    float s2 = dx * dx;
#pragma unroll
    for (int o = 16; o > 0; o >>= 1) s2 += __shfl_xor(s2, o, 32);
    if ((t & 31) == 0) red1[t >> 5] = s2;
    __syncthreads();
    const float var = (red1[0] + red1[1] + red1[2] + red1[3]) * (1.0f / D);

    float y = dx * rsqrtf(var + 1e-5f) * g[t] + b[t];
    h[(size_t)node * D + t] = fmaxf(y, 0.0f);
}

// ---------------------------------------------------------------------------
// Launch: memset msg -> scatter -> gemm(L0) -> LN+ReLU -> memset -> scatter
//         -> gemm(L1) into d_out. All graph-capture safe.
// ---------------------------------------------------------------------------
extern "C" void kernel_launch(void* const* d_in, const int* in_sizes, int n_in,
                              void* d_out, int out_size, void* d_ws, size_t ws_size,
                              hipStream_t stream) {
    const float* feat    = (const float*)d_in[0];
    const float* W_self0 = (const float*)d_in[1];
    const float* W_nei0  = (const float*)d_in[2];
    const float* b0      = (const float*)d_in[3];
    const float* W_self1 = (const float*)d_in[4];
    const float* W_nei1  = (const float*)d_in[5];
    const float* b1      = (const float*)d_in[6];
    const float* ln_g    = (const float*)d_in[7];
    const float* ln_b    = (const float*)d_in[8];
    const int*   esrc    = (const int*)d_in[9];
    const int*   edst    = (const int*)d_in[10];
    const int*   indeg   = (const int*)d_in[11];
    float* out = (float*)d_out;

    const size_t ND = (size_t)N_NODES * D;   // 12.8M floats = 51.2 MB
    float* msg = (float*)d_ws;               // [0, ND)
    float* h   = msg + ND;                   // [ND, 2*ND)

    const int scat_blocks = (N_EDGES * 128) / 256;   // 800000
    const int gemm_blocks = N_NODES / 16;            // 6250

    // ---- layer 0 ----
    hipMemsetAsync(msg, 0, ND * sizeof(float), stream);
    sage_scatter<<<scat_blocks, 256, 0, stream>>>(feat, esrc, edst, msg);
    sage_gemm<<<gemm_blocks, 256, 0, stream>>>(feat, msg, W_self0, W_nei0, b0,
                                               indeg, h);
    ln_relu<<<N_NODES, 128, 0, stream>>>(h, ln_g, ln_b);

    // ---- layer 1 ----
    hipMemsetAsync(msg, 0, ND * sizeof(float), stream);
    sage_scatter<<<scat_blocks, 256, 0, stream>>>(h, esrc, edst, msg);
    sage_gemm<<<gemm_blocks, 256, 0, stream>>>(h, msg, W_self1, W_nei1, b1,
                                               indeg, out);
    (void)in_sizes; (void)n_in; (void)out_size; (void)ws_size;
}